// AttentionLayer_50989851738889
// MI455X (gfx1250) — compile-verified
//
#include <hip/hip_runtime.h>

typedef __attribute__((ext_vector_type(2))) float v2f;
typedef __attribute__((ext_vector_type(8))) float v8f;

#define BB 4
#define LL 1024
#define DD 512
#define UU 32
#define MTOT (BB * LL)      // 4096 rows of X
#define CHUNK 256           // K rows staged in LDS per iteration
#define KSTR 33             // padded LDS row stride (conflict-free for wave32)

// Native CDNA5 tanh (TRANS32 op). Prefer the builtin so the compiler handles
// the TRANS co-execution hazard; else inline asm + trailing v_nop (ISA 7.4:
// one independent op required after a TRANS op before its result is used).
__device__ __forceinline__ float fast_tanh(float x) {
#if __has_builtin(__builtin_amdgcn_tanhf)
    return __builtin_amdgcn_tanhf(x);
#elif __has_builtin(__builtin_amdgcn_tanh_f32)
    return __builtin_amdgcn_tanh_f32(x);
#else
    float r;
    asm volatile("v_tanh_f32 %0, %1\n\tv_nop" : "=v"(r) : "v"(x));
    return r;
#endif
}

// ---------------------------------------------------------------------------
// Kernel 1: Q = X*Wt and K = X*Wx with V_WMMA_F32_16X16X4_F32 (exact fp32).
// grid.x = MTOT/16 row tiles. block = 128 threads = 4 waves:
//   wave0: Q cols 0-15, wave1: Q cols 16-31, wave2: K cols 0-15, wave3: K cols 16-31
// ---------------------------------------------------------------------------
__global__ __launch_bounds__(128)
void qk_wmma_kernel(const float* __restrict__ X,
                    const float* __restrict__ Wt,
                    const float* __restrict__ Wx,
                    float* __restrict__ Q,
                    float* __restrict__ K)
{
    const int lane = threadIdx.x & 31;
    const int wave = threadIdx.x >> 5;
    const int m0   = blockIdx.x * 16;
    const int n0   = (wave & 1) * 16;
    const float* __restrict__ W   = (wave & 2) ? Wx : Wt;
    float*       __restrict__ Out = (wave & 2) ? K  : Q;

    const int row   = m0 + (lane & 15);
    const int klane = (lane >> 4) * 2;      // 0 or 2
    const int ncol  = n0 + (lane & 15);

    const float* __restrict__ xrow = X + (size_t)row * DD;

    v8f acc = {};
    for (int kk = 0; kk < DD; kk += 4) {
        v2f a, b;
        a.x = xrow[kk + klane];
        a.y = xrow[kk + klane + 1];
        b.x = W[(size_t)(kk + klane) * UU + ncol];
        b.y = W[(size_t)(kk + klane + 1) * UU + ncol];
        acc = __builtin_amdgcn_wmma_f32_16x16x4_f32(
            /*neg_a=*/false, a, /*neg_b=*/false, b,
            /*c_mod=*/(short)0, acc, /*reuse_a=*/false, /*reuse_b=*/false);
    }

#pragma unroll
    for (int v = 0; v < 8; ++v) {
        const int r = m0 + v + ((lane >> 4) ? 8 : 0);
        Out[(size_t)r * UU + ncol] = acc[v];
    }
}

// ---------------------------------------------------------------------------
// Kernel 2: per (b,s) row: gmax = max_t sum_u tanh(q[s,u]+k[t,u]+bg[u])*Wa[u]
// One block of 256 threads per row. K rows staged through LDS in CHUNK tiles
// (coalesced float4 fill, padded stride 33 -> conflict-free wave32 reads).
// Native v_tanh_f32 for the 134M-tanh hot loop; prefetch next tile into L2.
// ---------------------------------------------------------------------------
__global__ __launch_bounds__(256)
void score_max_kernel(const float* __restrict__ Q,
                      const float* __restrict__ Km,
                      const float* __restrict__ bg,
                      const float* __restrict__ Wa,
                      float* __restrict__ Gmax)
{
    __shared__ float sK[CHUNK * KSTR];   // ~33.8 KB of the 320 KB WGP LDS
    __shared__ float sC[UU];
    __shared__ float sW[UU];
    __shared__ float sRed[8];

    const int r    = blockIdx.x;         // 0..MTOT-1
    const int base = (r >> 10) << 10;    // first row of this batch in K
    const int tid  = threadIdx.x;

    if (tid < UU) {
        sC[tid] = Q[(size_t)r * UU + tid] + bg[tid];
        sW[tid] = Wa[tid];
    }
    __syncthreads();

    float c[UU], w[UU];
#pragma unroll
    for (int u = 0; u < UU; ++u) { c[u] = sC[u]; w[u] = sW[u]; }

    float gmax = -1e30f;
    for (int t0 = 0; t0 < LL; t0 += CHUNK) {
        // coalesced cooperative fill: 2048 float4s, 8 per thread
        const float4* __restrict__ src =
            (const float4*)(Km + (size_t)(base + t0) * UU);
#pragma unroll
        for (int it = 0; it < (CHUNK * (UU / 4)) / 256; ++it) {
            const int f    = tid + it * 256;
            const float4 v = src[f];
            float* dst = sK + (f >> 3) * KSTR + ((f & 7) << 2);
            dst[0] = v.x; dst[1] = v.y; dst[2] = v.z; dst[3] = v.w;
        }
        __syncthreads();

        // warm the next 32KB tile while tanh work runs (global_prefetch_b8);
        // each thread touches one distinct 128B line
        if (t0 + CHUNK < LL) {
            const char* nxt =
                (const char*)(Km + (size_t)(base + t0 + CHUNK) * UU) +
                (size_t)tid * 128;
            __builtin_prefetch(nxt, 0, 0);
        }

        // thread tid evaluates t = t0 + tid (LDS reads conflict-free)
        const float* __restrict__ krow = sK + tid * KSTR;
        float g = 0.f;
#pragma unroll
        for (int u = 0; u < UU; ++u)
            g = fmaf(fast_tanh(c[u] + krow[u]), w[u], g);
        gmax = fmaxf(gmax, g);
        __syncthreads();
    }

    // wave32 max reduction, then cross-wave via LDS
#pragma unroll
    for (int off = 16; off >= 1; off >>= 1)
        gmax = fmaxf(gmax, __shfl_xor(gmax, off, 32));
    if ((tid & 31) == 0) sRed[tid >> 5] = gmax;
    __syncthreads();
    if (tid == 0) {
        float m = sRed[0];
#pragma unroll
        for (int i = 1; i < 8; ++i) m = fmaxf(m, sRed[i]);
        Gmax[r] = m;
    }
}

// ---------------------------------------------------------------------------
// Kernel 3: e = exp(gmax); a = e / e (faithful to reference; == 1 unless NaN);
// out = a * x. float4-vectorized: the only HBM-bound stage (16 MB total).
// ---------------------------------------------------------------------------
__global__ __launch_bounds__(256)
void finalize_kernel(const float* __restrict__ X,
                     const float* __restrict__ Gmax,
                     float* __restrict__ Out)
{
    const int idx  = blockIdx.x * 256 + threadIdx.x;  // float4 index
    const int rowv = DD / 4;                          // 128 float4 per row
    const int r    = idx / rowv;
    const float e  = expf(Gmax[r]);
    const float a  = e / e;
    const float4 x = ((const float4*)X)[idx];
    float4 o;
    o.x = a * x.x; o.y = a * x.y; o.z = a * x.z; o.w = a * x.w;
    ((float4*)Out)[idx] = o;
}

// ---------------------------------------------------------------------------
extern "C" void kernel_launch(void* const* d_in, const int* in_sizes, int n_in,
                              void* d_out, int out_size, void* d_ws, size_t ws_size,
                              hipStream_t stream) {
    const float* X  = (const float*)d_in[0];   // [B, L, D]
    const float* Wt = (const float*)d_in[1];   // [D, U]
    const float* Wx = (const float*)d_in[2];   // [D, U]
    const float* bg = (const float*)d_in[3];   // [U]
    const float* Wa = (const float*)d_in[4];   // [U, 1]
    float* Out = (float*)d_out;                // [B, L, D]

    char* ws = (char*)d_ws;
    float* Q    = (float*)(ws);                               // 4096*32 f32
    float* K    = (float*)(ws + (size_t)MTOT * UU * 4);       // 4096*32 f32
    float* Gmax = (float*)(ws + (size_t)2 * MTOT * UU * 4);   // 4096 f32

    qk_wmma_kernel<<<MTOT / 16, 128, 0, stream>>>(X, Wt, Wx, Q, K);
    score_max_kernel<<<MTOT, 256, 0, stream>>>(Q, K, bg, Wa, Gmax);
    finalize_kernel<<<(MTOT * DD / 4) / 256, 256, 0, stream>>>(X, Gmax, Out);
}